// GCNLayer_74010876444909
// MI455X (gfx1250) — compile-verified
//
#include <hip/hip_runtime.h>
#include <math.h>

#define N_NODES 100000
#define D 128

typedef __attribute__((ext_vector_type(2))) float v2f;
typedef __attribute__((ext_vector_type(8))) float v8f;

// ---------------------------------------------------------------------------
// GEMM: h[m][n] = sum_k x[m][k] * W[n][k]   (h = x @ W^T), fp32 WMMA 16x16x4.
// One wave computes a 16-row x 128-col stripe (8 C tiles), K-loop over 128.
// Codegen verified round 1: 32 v_wmma per unrolled iter, clause'd b64 loads.
// ---------------------------------------------------------------------------
__global__ __launch_bounds__(128) void gcn_gemm_kernel(
    const float* __restrict__ x, const float* __restrict__ W,
    float* __restrict__ h) {
  const int lane     = threadIdx.x & 31;
  const int waveId   = threadIdx.x >> 5;
  const int row_tile = blockIdx.x * 4 + waveId;       // 16 rows per tile
  if (row_tile * 16 >= N_NODES) return;               // wave-uniform guard
  const int row0 = row_tile * 16;

  // A fragment addressing (16x4 f32: lanes 0-15 -> K=0,1; lanes 16-31 -> K=2,3)
  const int a_row  = row0 + (lane & 15);
  const int koff   = (lane >> 4) * 2;                 // 0 or 2
  const int n_lane = lane & 15;

  v8f c[8];
#pragma unroll
  for (int t = 0; t < 8; ++t) c[t] = (v8f){0.f,0.f,0.f,0.f,0.f,0.f,0.f,0.f};

  for (int k = 0; k < D; k += 4) {
    const v2f a = *(const v2f*)(x + (size_t)a_row * D + k + koff);
#pragma unroll
    for (int t = 0; t < 8; ++t) {
      const int n = t * 16 + n_lane;                  // B[k][n] = W[n][k]
      const v2f b = *(const v2f*)(W + (size_t)n * D + k + koff);
      c[t] = __builtin_amdgcn_wmma_f32_16x16x4_f32(
          /*neg_a=*/false, a, /*neg_b=*/false, b,
          /*c_mod=*/(short)0, c[t], /*reuse_a=*/false, /*reuse_b=*/false);
    }
  }

  // C/D layout: VGPR i -> M = m_base + i (m_base = 8 for lanes 16-31), N = lane%16
  const int m_base = (lane >> 4) * 8;
#pragma unroll
  for (int t = 0; t < 8; ++t) {
#pragma unroll
    for (int i = 0; i < 8; ++i) {
      h[(size_t)(row0 + m_base + i) * D + t * 16 + n_lane] = c[t][i];
    }
  }
}

// ---------------------------------------------------------------------------
// SpMM scatter: one edge per wave32; lane l handles 4 consecutive features.
// h (51.2MB) and out (51.2MB) both fit in 192MB L2: gathers + atomics resolve
// at L2. Edge metadata index goes through readfirstlane so the three metadata
// reads become s_load_b32 (K$) instead of 32-lane-replicated vector loads.
// The fp32 adds are forced to native global_atomic_add_f32 (no CAS fallback).
// ---------------------------------------------------------------------------
__device__ __forceinline__ void atomic_add_f32_native(float* addr, float val) {
  asm volatile("global_atomic_add_f32 %0, %1, off"
               :
               : "v"(addr), "v"(val)
               : "memory");
}

__global__ __launch_bounds__(256) void gcn_spmm_kernel(
    const float* __restrict__ h, const float* __restrict__ vals,
    const int* __restrict__ src, const int* __restrict__ dst,
    float* __restrict__ out, int n_edges) {
  const int lane = threadIdx.x & 31;
  const int e0   = blockIdx.x * 8 + (threadIdx.x >> 5);
  if (e0 >= n_edges) return;                          // wave-uniform guard
  const int e = __builtin_amdgcn_readfirstlane(e0);   // provably uniform
  const float v = vals[e];                            // -> s_load_b32
  const int   s = src[e];                             // -> s_load_b32
  const int   d = dst[e];                             // -> s_load_b32
  const float4 hv = *(const float4*)(h + (size_t)s * D + lane * 4);
  float* o = out + (size_t)d * D + lane * 4;
  atomic_add_f32_native(o + 0, hv.x * v);
  atomic_add_f32_native(o + 1, hv.y * v);
  atomic_add_f32_native(o + 2, hv.z * v);
  atomic_add_f32_native(o + 3, hv.w * v);
}

// ---------------------------------------------------------------------------
// Exact GELU in-place, float4-vectorized: 0.5*x*(1+erf(x/sqrt(2)))
// n is guaranteed divisible by 4 (N_NODES*128).
// ---------------------------------------------------------------------------
__device__ __forceinline__ float gelu1(float a) {
  return 0.5f * a * (1.0f + erff(a * 0.70710678118654752440f));
}

__global__ __launch_bounds__(256) void gcn_gelu_kernel(float4* __restrict__ out,
                                                       int n4) {
  const int i = blockIdx.x * blockDim.x + threadIdx.x;
  if (i < n4) {
    float4 a = out[i];                                // global_load_b128
    a.x = gelu1(a.x);
    a.y = gelu1(a.y);
    a.z = gelu1(a.z);
    a.w = gelu1(a.w);
    out[i] = a;                                       // global_store_b128
  }
}

extern "C" void kernel_launch(void* const* d_in, const int* in_sizes, int n_in,
                              void* d_out, int out_size, void* d_ws, size_t ws_size,
                              hipStream_t stream) {
  const float* x    = (const float*)d_in[0];   // [N_NODES, D]
  const float* W    = (const float*)d_in[1];   // [D, D]
  const float* vals = (const float*)d_in[2];   // [E]
  const int*   src  = (const int*)d_in[3];     // [E]
  const int*   dst  = (const int*)d_in[4];     // [E]
  float*       out  = (float*)d_out;           // [N_NODES, D]
  float*       h    = (float*)d_ws;            // scratch: [N_NODES, D] = 51.2 MB

  const int n_edges = in_sizes[2];

  // zero the accumulator (graph-capture-safe async memset)
  hipMemsetAsync(d_out, 0, (size_t)out_size * sizeof(float), stream);

  // 1) h = x @ W^T via fp32 WMMA
  const int row_tiles = (N_NODES + 15) / 16;          // 6250
  gcn_gemm_kernel<<<(row_tiles + 3) / 4, 128, 0, stream>>>(x, W, h);

  // 2) scatter-add: one wave per edge, native L2 fp32 atomics
  gcn_spmm_kernel<<<(n_edges + 7) / 8, 256, 0, stream>>>(h, vals, src, dst,
                                                         out, n_edges);

  // 3) exact GELU in-place (float4 lanes)
  const int n4 = (N_NODES * D) / 4;
  gcn_gelu_kernel<<<(n4 + 255) / 256, 256, 0, stream>>>((float4*)out, n4);
}